// s2cnn_small_regression_75505525063966
// MI455X (gfx1250) — compile-verified
//
#include <hip/hip_runtime.h>
#include <math.h>
#include <stdint.h>

// =====================================================================
// S2CNN pipeline for MI455X (gfx1250, wave32, WMMA).
//
// Heavy math = per-degree complex GEMMs -> V_WMMA_F32_16X16X4_F32.
// B tiles staged into LDS with gfx1250 async LDS DMA
// (GLOBAL_LOAD_ASYNC_TO_LDS_B128 + s_wait_asynccnt).
// Working set (~166MB) fits in 192MB L2 -> compute-bound -> fp32 WMMA.
// All reference constants (Wigner-d, quadrature, SO(3) basis) are
// recomputed on-device each call (deterministic).
// =====================================================================

#define PI_D 3.14159265358979323846
#define PI_F 3.14159265358979323846f

typedef float v2f __attribute__((ext_vector_type(2)));
typedef float v8f __attribute__((ext_vector_type(8)));

// ---------------- workspace layout (float offsets) -------------------
static constexpr long O_D16  = 0;                 // d^l(beta), 32 betas, l<16 : 32*5456
static constexpr long O_D8   = O_D16 + 174592;    // d^l(beta), 16 betas, l<8  : 16*680
static constexpr long O_DC32 = O_D8  + 10880;     // d^l_{m,0}, 64 betas, l<16 : 64*256
static constexpr long O_W32  = O_DC32 + 16384;    // 64
static constexpr long O_W16  = O_W32 + 64;        // 32
static constexpr long O_W8   = O_W16 + 32;        // 16
static constexpr long O_BS2  = O_W8  + 16;        // S2 basis  float2 x 256*6
static constexpr long O_B1   = O_BS2 + 3072;      // SO3 basis float2 x 5456*36
static constexpr long O_B2   = O_B1  + 392832;    // SO3 basis float2 x 680*36
static constexpr long O_B3   = O_B2  + 48960;
static constexpr long O_K2S  = O_B3  + 48960;     // scaled k2: 64*64*6
static constexpr long P_XS   = O_K2S + 24576;     // spectral in  (2 planes, max 4*5456*64)
static constexpr long P_ZL   = P_XS  + 2793472;   // spectral out (2 planes, max 4*5456*128)
static constexpr long P_PSI  = P_ZL  + 5586944;   // psi / mh / u / h2 / h4 (15.7M floats)
static constexpr long P_HA   = P_PSI + 15745024;  // h1 / h3 / h5 (16.8M floats)
// total ~ 41.6M floats ~ 166 MB

// ---------------- Wigner small-d (double, log-gamma stable) ----------
__device__ inline double wig_d(int l, int mp, int m, double beta)
{
  double ch = cos(0.5 * beta), sh = sin(0.5 * beta);
  int smin = m - mp; if (smin < 0) smin = 0;
  int smax = l + m;  if (l - mp < smax) smax = l - mp;
  double pref = 0.5 * (lgamma((double)(l + m) + 1.0) + lgamma((double)(l - m) + 1.0)
                     + lgamma((double)(l + mp) + 1.0) + lgamma((double)(l - mp) + 1.0));
  double acc = 0.0;
  for (int s = smin; s <= smax; ++s) {
    double ln = pref - (lgamma((double)(l + m - s) + 1.0) + lgamma((double)s + 1.0)
                      + lgamma((double)(mp - m + s) + 1.0) + lgamma((double)(l - mp - s) + 1.0));
    double t = exp(ln) * pow(ch, (double)(2 * l + m - mp - 2 * s))
                       * pow(sh, (double)(mp - m + 2 * s));
    acc += ((mp - m + s) & 1) ? -t : t;
  }
  return acc;
}

// ---------------- constant-table fill kernels ------------------------
__global__ void fill_qweights(float* __restrict__ out, int bq)
{
  int j = blockIdx.x * blockDim.x + threadIdx.x;
  if (j >= 2 * bq) return;
  double s = 0.0;
  for (int k = 0; k < bq; ++k)
    s += sin(PI_D * (2 * j + 1) * (2 * k + 1) / (4.0 * bq)) / (2 * k + 1);
  out[j] = (float)(2.0 / bq * sin(PI_D * (2 * j + 1) / (4.0 * bq)) * s * (2.0 * PI_D / (2 * bq)));
}

__global__ void fill_wigner_full(float* __restrict__ out, int nb, int Lspec)
{
  long total = (long)nb * ((long)Lspec * (4 * Lspec * Lspec - 1) / 3);
  for (long idx = (long)blockIdx.x * blockDim.x + threadIdx.x; idx < total;
       idx += (long)gridDim.x * blockDim.x) {
    int l = 0; long base = 0;
    for (;;) { long sz = (long)nb * (2 * l + 1) * (2 * l + 1);
               if (idx < base + sz) break; base += sz; ++l; }
    long rem = idx - base;
    int P = 2 * l + 1;
    int b = (int)(rem / (P * P));
    int ii = (int)(rem % (P * P));
    double beta = (b + 0.5) * PI_D / nb;
    out[idx] = (float)wig_d(l, ii / P - l, ii % P - l, beta);
  }
}

__global__ void fill_wigner_col(float* __restrict__ out, int nb, int Lspec) // d^l_{m,0}
{
  long total = (long)nb * Lspec * Lspec;
  for (long idx = (long)blockIdx.x * blockDim.x + threadIdx.x; idx < total;
       idx += (long)gridDim.x * blockDim.x) {
    int l = 0; long base = 0;
    for (;;) { long sz = (long)nb * (2 * l + 1);
               if (idx < base + sz) break; base += sz; ++l; }
    long rem = idx - base;
    int P = 2 * l + 1;
    int b = (int)(rem / P), mm = (int)(rem % P);
    double beta = (b + 0.5) * PI_D / nb;
    out[idx] = (float)wig_d(l, mm - l, 0, beta);
  }
}

__global__ void fill_basis_s2(float2* __restrict__ out)
{
  int idx = blockIdx.x * blockDim.x + threadIdx.x;
  if (idx >= 256 * 6) return;
  int p = idx % 6, slot = idx / 6;
  int l = 0; while ((l + 1) * (l + 1) <= slot) ++l;
  int m = slot - l * l - l;
  double alpha = p * (2.0 * PI_D / 6.0);
  double d = wig_d(l, m, 0, PI_D / 16.0);
  double phi = m * alpha;
  out[idx] = make_float2((float)(d * cos(phi)), (float)(-d * sin(phi)));
}

__global__ void fill_basis_so3(float2* __restrict__ out, int Lspec, double beta)
{
  long total = ((long)Lspec * (4 * Lspec * Lspec - 1) / 3) * 36;
  for (long idx = (long)blockIdx.x * blockDim.x + threadIdx.x; idx < total;
       idx += (long)gridDim.x * blockDim.x) {
    int p = (int)(idx % 36);
    long slot = idx / 36;
    int l = 0; long base = 0;
    for (;;) { long sz = (long)(2 * l + 1) * (2 * l + 1);
               if (slot < base + sz) break; base += sz; ++l; }
    int P = 2 * l + 1;
    int ii = (int)(slot - base);
    int m = ii / P - l, n = ii % P - l;
    double alpha = (p / 6) * (2.0 * PI_D / 6.0);
    double gamma = (p % 6) * (2.0 * PI_D / 6.0) - PI_D;
    double d = wig_d(l, m, n, beta);
    double phi = m * alpha + n * gamma;
    out[idx] = make_float2((float)(d * cos(phi)), (float)(-d * sin(phi)));
  }
}

__global__ void scale_copy(const float* __restrict__ in, float* __restrict__ out,
                           float s, int n)
{
  for (int i = blockIdx.x * blockDim.x + threadIdx.x; i < n; i += gridDim.x * blockDim.x)
    out[i] = in[i] * s;
}

// ---------------- stage kernels --------------------------------------
// conv1 (1x1 over channels) + relu : h1[z,o,beta,alpha]
__global__ void conv1_relu(const float* __restrict__ x, const float* __restrict__ w,
                           const float* __restrict__ bias, float* __restrict__ h)
{
  int total = 4 * 64 * 4096;
  for (int idx = blockIdx.x * blockDim.x + threadIdx.x; idx < total;
       idx += gridDim.x * blockDim.x) {
    int pix = idx & 4095; int t = idx >> 12; int o = t & 63; int z = t >> 6;
    float acc = bias[o];
    for (int c = 0; c < 4; ++c)
      acc += w[o * 4 + c] * x[((long)(z * 4 + c) << 12) + pix];
    h[idx] = acc > 0.f ? acc : 0.f;
  }
}

// DFT over alpha axis for m=-15..15 : mh[zc][b][mm31] (planar complex)
__global__ void dft_alpha(const float* __restrict__ h, float* __restrict__ mre,
                          float* __restrict__ mim)
{
  int total = 4 * 64 * 64 * 31;
  const float wga = 6.28318530717958647692f / 64.f;
  for (int idx = blockIdx.x * blockDim.x + threadIdx.x; idx < total;
       idx += gridDim.x * blockDim.x) {
    int mm = idx % 31; int t = idx / 31; int b = t & 63; int zc = t >> 6;
    int m = mm - 15;
    const float* row = h + ((long)zc * 64 + b) * 64;
    float ar = 0.f, ai = 0.f;
    for (int a = 0; a < 64; ++a) {
      int tt = (m * a) & 63;
      float sn, cs; __sincosf(wga * (float)tt, &sn, &cs);
      ar += row[a] * cs; ai -= row[a] * sn;
    }
    mre[idx] = ar; mim[idx] = ai;
  }
}

// S2 spectral: xs[z][slot=l^2+mm][c] (planar, plane 4*256*64)
__global__ void s2_spec(const float* __restrict__ mre, const float* __restrict__ mim,
                        const float* __restrict__ dcol, const float* __restrict__ w32,
                        float* __restrict__ xre, float* __restrict__ xim)
{
  int total = 4 * 256 * 64;
  for (int idx = blockIdx.x * blockDim.x + threadIdx.x; idx < total;
       idx += gridDim.x * blockDim.x) {
    int c = idx & 63; int t = idx >> 6; int slot = t & 255; int z = t >> 8;
    int l = 0; while ((l + 1) * (l + 1) <= slot) ++l;
    int mm = slot - l * l; int P = 2 * l + 1;
    int mm15 = (mm - l) + 15;
    const float* dp = dcol + 64 * l * l + mm;
    long mbase = ((long)(z * 64 + c) * 64) * 31 + mm15;
    float ar = 0.f, ai = 0.f;
    for (int b = 0; b < 64; ++b) {
      float dv = dp[b * P] * w32[b];
      ar += dv * mre[mbase + b * 31];
      ai += dv * mim[mbase + b * 31];
    }
    long a = (long)z * 256 * 64 + (long)slot * 64 + c;
    xre[a] = ar; xim[a] = ai;
  }
}

// ---------------- THE WMMA GEMM (fp32 16x16x4) ------------------------
// C(complex) = A(complex, planar) x B(complex or real, planar, row-major KxN)
// Requirements guaranteed by all call sites: K % 16 == 0, N % 128 == 0.
// B tiles (16 x 128, both planes) are staged into LDS with the gfx1250
// async LDS DMA (GLOBAL_LOAD_ASYNC_TO_LDS_B128, ASYNCcnt-tracked), then
// WMMA fragments are fed from LDS and shared by all 8 waves of the block.
__global__ __launch_bounds__(256)
void gemm_wmma_cplx(const float* __restrict__ Are, const float* __restrict__ Aim,
                    const float* __restrict__ Bre, const float* __restrict__ Bim,
                    float* __restrict__ Cre, float* __restrict__ Cim,
                    int M, int N, int K,
                    int rowsPerZ, int AzStr, int CzStr, int CrowStr)
{
  __shared__ __align__(16) float sBre[16 * 128];
  __shared__ __align__(16) float sBim[16 * 128];
  const int tid  = threadIdx.x;
  const int lane = tid & 31;
  const int wave = tid >> 5;
  const int colBase = blockIdx.y * 128;       // N % 128 == 0 -> always in range
  const int wcol = wave * 16 + (lane & 15);
  const int col  = colBase + wcol;
  const int rA   = blockIdx.x * 16 + (lane & 15);
  const int k0   = (lane & 16) ? 2 : 0;
  const bool aOk = rA < M;
  long aBase = 0;
  if (aOk) {
    int z = rA / rowsPerZ, mr = rA % rowsPerZ;
    aBase = (long)z * AzStr + (long)mr * K;
  }
  v8f cr = {0.f,0.f,0.f,0.f,0.f,0.f,0.f,0.f};
  v8f ci = {0.f,0.f,0.f,0.f,0.f,0.f,0.f,0.f};
  const bool hasBi = (Bim != nullptr);

  for (int kk0 = 0; kk0 < K; kk0 += 16) {
    __syncthreads();   // previous block's LDS reads complete before overwrite
    // ---- async stage of B tile: 2048 floats/plane, 16B per lane per op ----
    for (int t = 0; t < 2; ++t) {
      int e4 = tid + t * 256;                 // vector id in [0,512)
      int krow = e4 >> 5, c4 = (e4 & 31) << 2;
      unsigned loff = (unsigned)(uintptr_t)&sBre[krow * 128 + c4];   // low 32b = LDS addr
      unsigned long long g =
          (unsigned long long)(uintptr_t)(Bre + (long)(kk0 + krow) * N + colBase + c4);
      asm volatile("global_load_async_to_lds_b128 %0, %1, off"
                   :: "v"(loff), "v"(g) : "memory");
    }
    if (hasBi) {
      for (int t = 0; t < 2; ++t) {
        int e4 = tid + t * 256;
        int krow = e4 >> 5, c4 = (e4 & 31) << 2;
        unsigned loff = (unsigned)(uintptr_t)&sBim[krow * 128 + c4];
        unsigned long long g =
            (unsigned long long)(uintptr_t)(Bim + (long)(kk0 + krow) * N + colBase + c4);
        asm volatile("global_load_async_to_lds_b128 %0, %1, off"
                     :: "v"(loff), "v"(g) : "memory");
      }
    }
    if (aOk && kk0 + 16 < K) __builtin_prefetch(&Are[aBase + kk0 + 16], 0, 1);
#if defined(__has_builtin) && __has_builtin(__builtin_amdgcn_s_wait_asynccnt)
    __builtin_amdgcn_s_wait_asynccnt(0);
#else
    asm volatile("s_wait_asynccnt 0x0" ::: "memory");
#endif
    __syncthreads();   // all waves' DMA done -> tile visible block-wide
    // ---- 4 WMMA K-steps out of the staged tile ----
#pragma unroll
    for (int kq = 0; kq < 16; kq += 4) {
      const long ka = kk0 + kq + k0;
      v2f ar = {0.f, 0.f}, ai = {0.f, 0.f};
      if (aOk) {
        ar.x = Are[aBase + ka];  ar.y = Are[aBase + ka + 1];
        ai.x = Aim[aBase + ka];  ai.y = Aim[aBase + ka + 1];
      }
      const int lrow = kq + k0;
      v2f br; br.x = sBre[lrow * 128 + wcol]; br.y = sBre[(lrow + 1) * 128 + wcol];
      cr = __builtin_amdgcn_wmma_f32_16x16x4_f32(false, ar, false, br, (short)0, cr, false, false);
      ci = __builtin_amdgcn_wmma_f32_16x16x4_f32(false, ai, false, br, (short)0, ci, false, false);
      if (hasBi) {
        v2f bi; bi.x = sBim[lrow * 128 + wcol]; bi.y = sBim[(lrow + 1) * 128 + wcol];
        ci = __builtin_amdgcn_wmma_f32_16x16x4_f32(false, ar, false, bi, (short)0, ci, false, false);
        v2f bin = { -bi.x, -bi.y };
        cr = __builtin_amdgcn_wmma_f32_16x16x4_f32(false, ai, false, bin, (short)0, cr, false, false);
      }
    }
  }
  const int rowHi = (lane & 16) ? 8 : 0;
  for (int j = 0; j < 8; ++j) {
    int r = blockIdx.x * 16 + rowHi + j;
    if (r < M) {
      int z = r / rowsPerZ, mr = r % rowsPerZ;
      long ca = (long)z * CzStr + (long)mr * CrowStr + col;
      Cre[ca] = cr[j];
      Cim[ca] = ci[j];
    }
  }
}

// S2 layer: zl[z][off2+mm*P+nn][o] = sum_p u[z,mm,(o,p)] * conj(bas_s2[nn,p])
__global__ void s2_zl(const float* __restrict__ Ure, const float* __restrict__ Uim,
                      const float2* __restrict__ bas,
                      float* __restrict__ Zre, float* __restrict__ Zim,
                      int P, int Qtot, int Co, int off2)
{
  int total = 4 * P * P * Co;
  for (int idx = blockIdx.x * blockDim.x + threadIdx.x; idx < total;
       idx += gridDim.x * blockDim.x) {
    int o = idx % Co; int t = idx / Co;
    int nn = t % P; t /= P; int mm = t % P; int z = t / P;
    const float2* bp = bas + nn * 6;
    const float* ur = Ure + ((long)z * P + mm) * 384 + o * 6;
    const float* ui = Uim + ((long)z * P + mm) * 384 + o * 6;
    float re = 0.f, im = 0.f;
    for (int p = 0; p < 6; ++p) {
      float2 b = bp[p];
      re += ur[p] * b.x + ui[p] * b.y;   // u * conj(b)
      im += ui[p] * b.x - ur[p] * b.y;
    }
    long a = (long)z * Qtot * Co + (long)(off2 + mm * P + nn) * Co + o;
    Zre[a] = re; Zim[a] = im;
  }
}

// psi (conj, scaled) in GEMM-B layout: B[(k*Ci+i)][(n*Co+o)] = sum_p k[i,o,p]*SC*conj(bas[n,k,p])
__global__ void psi_build(const float* __restrict__ kern, const float2* __restrict__ bas,
                          float scale, float* __restrict__ Bre, float* __restrict__ Bim,
                          int P, int Ci, int Co)
{
  long total = (long)P * Ci * P * Co;
  int N = P * Co;
  for (long idx = (long)blockIdx.x * blockDim.x + threadIdx.x; idx < total;
       idx += (long)gridDim.x * blockDim.x) {
    int col = (int)(idx % N); long rowK = idx / N;
    int i = (int)(rowK % Ci); int k = (int)(rowK / Ci);
    int o = col % Co; int nn = col / Co;
    const float*  kp = kern + ((long)i * Co + o) * 36;
    const float2* bp = bas + ((long)nn * P + k) * 36;
    float re = 0.f, im = 0.f;
    for (int p = 0; p < 36; ++p) {
      float kv = kp[p];
      float2 bv = bp[p];
      re += kv * bv.x; im -= kv * bv.y;
    }
    Bre[idx] = re * scale; Bim[idx] = im * scale;
  }
}

// SO(3) inverse transform + bias + relu: one block per (z,f,b) slice
__global__ __launch_bounds__(256)
void so3_ifft_bias_relu(const float* __restrict__ zre, const float* __restrict__ zim,
                        const float* __restrict__ dtab, const float* __restrict__ bias,
                        float* __restrict__ out, int n, int Lspec, int Qspec, int Cch)
{
  __shared__ float Tre[1024], Tim[1024], Gre[1024], Gim[1024];
  int bidx = blockIdx.x;
  int b = bidx % n; int zf = bidx / n; int f = zf % Cch; int z = zf / Cch;
  int n2 = n * n;
  int Lmax = Lspec - 1;
  const float wga = 6.28318530717958647692f / n;
  for (int e = threadIdx.x; e < n2; e += 256) {
    int r = e / n, c = e % n;
    int mr = (r <= Lmax) ? r : (r - n);
    int mc = (c <= Lmax) ? c : (c - n);
    float accr = 0.f, acci = 0.f;
    if (mr >= -Lmax && mr <= Lmax && mc >= -Lmax && mc <= Lmax) {
      int am = mr < 0 ? -mr : mr, ac = mc < 0 ? -mc : mc;
      int lmin = am > ac ? am : ac;
      for (int l = lmin; l <= Lmax; ++l) {
        int P = 2 * l + 1;
        int o2 = l * (4 * l * l - 1) / 3;
        int ii = (mr + l) * P + (mc + l);
        float dv = dtab[(long)n * o2 + (long)b * P * P + ii] * (float)P;
        long za = (long)z * Qspec * Cch + (long)(o2 + ii) * Cch + f;
        accr += dv * zre[za];
        acci += dv * zim[za];
      }
    }
    Tre[e] = accr; Tim[e] = acci;
  }
  __syncthreads();
  for (int e = threadIdx.x; e < n2; e += 256) {   // G[r,g] = sum_c T[r,c] e^{+i2pi cg/n}
    int r = e / n, g = e % n;
    float ar = 0.f, ai = 0.f;
    for (int c = 0; c < n; ++c) {
      int t = (c * g) & (n - 1);
      float sn, cs; __sincosf(wga * (float)t, &sn, &cs);
      float tr = Tre[r * n + c], ti = Tim[r * n + c];
      ar += tr * cs - ti * sn;
      ai += tr * sn + ti * cs;
    }
    Gre[e] = ar; Gim[e] = ai;
  }
  __syncthreads();
  float bv = bias[f];
  for (int e = threadIdx.x; e < n2; e += 256) {   // out[a,g] = Re sum_r G[r,g] e^{+i2pi ra/n}
    int a = e / n, g = e % n;
    float v = 0.f;
    for (int r = 0; r < n; ++r) {
      int t = (r * a) & (n - 1);
      float sn, cs; __sincosf(wga * (float)t, &sn, &cs);
      v += Gre[r * n + g] * cs - Gim[r * n + g] * sn;
    }
    v += bv;
    out[(long)bidx * n2 + e] = v > 0.f ? v : 0.f;
  }
}

// SO(3) forward transform: one block per (z,c), loops beta slices in LDS
__global__ __launch_bounds__(256)
void so3_spec(const float* __restrict__ x, const float* __restrict__ dtab,
              const float* __restrict__ wq,
              float* __restrict__ xre, float* __restrict__ xim,
              int n, int Lspec, int Qspec, int Cch)
{
  __shared__ float sx[1024];
  __shared__ float Hre[1024], Him[1024];
  __shared__ float Fre[961], Fim[961];
  __shared__ float accR[5456], accI[5456];
  int zc = blockIdx.x; int c = zc % Cch; int z = zc / Cch;
  int Lmax = Lspec - 1; int W = 2 * Lmax + 1;
  int n2 = n * n;
  const float wga = 6.28318530717958647692f / n;
  for (int s = threadIdx.x; s < Qspec; s += 256) { accR[s] = 0.f; accI[s] = 0.f; }
  __syncthreads();
  for (int b = 0; b < n; ++b) {
    for (int e = threadIdx.x; e < n2; e += 256)
      sx[e] = x[((long)zc * n + b) * n2 + e];
    __syncthreads();
    for (int e = threadIdx.x; e < W * n; e += 256) {   // H[mm,g] = sum_a sx e^{-i2pi ma/n}
      int mm = e / n, g = e % n; int m = mm - Lmax;
      float ar = 0.f, ai = 0.f;
      for (int a = 0; a < n; ++a) {
        int t = (m * a) & (n - 1);
        float sn, cs; __sincosf(wga * (float)t, &sn, &cs);
        float v = sx[a * n + g];
        ar += v * cs; ai -= v * sn;
      }
      Hre[e] = ar; Him[e] = ai;
    }
    __syncthreads();
    for (int e = threadIdx.x; e < W * W; e += 256) {   // F[mm,kk] = sum_g H e^{-i2pi kg/n}
      int mm = e / W, kk = e % W; int k = kk - Lmax;
      float ar = 0.f, ai = 0.f;
      for (int g = 0; g < n; ++g) {
        int t = (k * g) & (n - 1);
        float sn, cs; __sincosf(wga * (float)t, &sn, &cs);
        float hr = Hre[mm * n + g], hi = Him[mm * n + g];
        ar += hr * cs + hi * sn;
        ai += hi * cs - hr * sn;
      }
      Fre[e] = ar; Fim[e] = ai;
    }
    __syncthreads();
    float wb = wq[b] * wga;
    for (int s = threadIdx.x; s < Qspec; s += 256) {
      int l = 0;
      while ((l + 1) * (4 * (l + 1) * (l + 1) - 1) / 3 <= s) ++l;
      int o2 = l * (4 * l * l - 1) / 3;
      int P = 2 * l + 1;
      int ii = s - o2;
      int m = ii / P - l, k = ii % P - l;
      float dv = dtab[(long)n * o2 + (long)b * P * P + ii] * wb;
      int fi = (m + Lmax) * W + (k + Lmax);
      accR[s] += dv * Fre[fi];
      accI[s] += dv * Fim[fi];
    }
    __syncthreads();
  }
  for (int s = threadIdx.x; s < Qspec; s += 256) {
    long a = (long)z * Qspec * Cch + (long)s * Cch + c;
    xre[a] = accR[s]; xim[a] = accI[s];
  }
}

// integrate over SO(3) with beta quadrature + final linear
__global__ __launch_bounds__(256)
void integrate_linear(const float* __restrict__ h5, const float* __restrict__ w8,
                      const float* __restrict__ lw, const float* __restrict__ lb,
                      float* __restrict__ out)
{
  __shared__ float red[256];
  int z = blockIdx.x; int f = threadIdx.x;
  const float scale = (6.28318530717958647692f / 16.f) * (6.28318530717958647692f / 16.f);
  float s = 0.f;
  const float* base = h5 + ((long)(z * 256 + f) * 16) * 256;
  for (int b = 0; b < 16; ++b) {
    float sb = 0.f;
    for (int ag = 0; ag < 256; ++ag) sb += base[b * 256 + ag];
    s += sb * w8[b];
  }
  red[f] = s * scale * lw[f];
  __syncthreads();
  for (int st = 128; st > 0; st >>= 1) {
    if (f < st) red[f] += red[f + st];
    __syncthreads();
  }
  if (f == 0) out[z] = red[0] + lb[0];
}

// =====================================================================
extern "C" void kernel_launch(void* const* d_in, const int* in_sizes, int n_in,
                              void* d_out, int out_size, void* d_ws, size_t ws_size,
                              hipStream_t stream)
{
  (void)in_sizes; (void)n_in; (void)out_size; (void)ws_size;
  float* ws = (float*)d_ws;
  const float* x  = (const float*)d_in[0];
  const float* w1 = (const float*)d_in[1];
  const float* b1 = (const float*)d_in[2];
  const float* k2 = (const float*)d_in[3];
  const float* b2 = (const float*)d_in[4];
  const float* k3 = (const float*)d_in[5];
  const float* b3 = (const float*)d_in[6];
  const float* k4 = (const float*)d_in[7];
  const float* b4 = (const float*)d_in[8];
  const float* k5 = (const float*)d_in[9];
  const float* b5 = (const float*)d_in[10];
  const float* lw = (const float*)d_in[11];
  const float* lb = (const float*)d_in[12];
  float* out = (float*)d_out;

  const float SC2 = (float)(1.0 / sqrt(24576.0));
  const float SC3 = (float)(1.0 / sqrt(2304.0));
  const float SC4 = (float)(1.0 / 24.0);
  const float SC5 = (float)(1.0 / sqrt(4608.0));
  auto off2 = [](int l) { return l * (4 * l * l - 1) / 3; };
  auto cdiv = [](long a, long b) { return (int)((a + b - 1) / b); };

  // ---- constant tables ----
  fill_qweights<<<1, 64, 0, stream>>>(ws + O_W32, 32);
  fill_qweights<<<1, 32, 0, stream>>>(ws + O_W16, 16);
  fill_qweights<<<1, 16, 0, stream>>>(ws + O_W8, 8);
  fill_wigner_full<<<cdiv(174592, 256), 256, 0, stream>>>(ws + O_D16, 32, 16);
  fill_wigner_full<<<cdiv(10880, 256), 256, 0, stream>>>(ws + O_D8, 16, 8);
  fill_wigner_col <<<cdiv(16384, 256), 256, 0, stream>>>(ws + O_DC32, 64, 16);
  fill_basis_s2  <<<cdiv(1536, 256), 256, 0, stream>>>((float2*)(ws + O_BS2));
  fill_basis_so3 <<<cdiv(196416, 256), 256, 0, stream>>>((float2*)(ws + O_B1), 16, PI_D / 16.0);
  fill_basis_so3 <<<cdiv(24480, 256), 256, 0, stream>>>((float2*)(ws + O_B2), 8, PI_D / 8.0);
  fill_basis_so3 <<<cdiv(24480, 256), 256, 0, stream>>>((float2*)(ws + O_B3), 8, PI_D / 4.0);
  scale_copy<<<cdiv(24576, 256), 256, 0, stream>>>(k2, ws + O_K2S, SC2, 24576);

  auto launch_gemm = [&](const float* Are, const float* Aim, const float* Bre,
                         const float* Bim, float* Cre, float* Cim,
                         int M, int N, int K, int rowsPerZ, int AzStr,
                         int CzStr, int CrowStr) {
    dim3 g((unsigned)((M + 15) / 16), (unsigned)(N / 128));   // N % 128 == 0 always
    gemm_wmma_cplx<<<g, 256, 0, stream>>>(Are, Aim, Bre, Bim, Cre, Cim,
                                          M, N, K, rowsPerZ, AzStr, CzStr, CrowStr);
  };

  // ---- stage 1: conv1 + relu -> h1 (HA) ----
  conv1_relu<<<cdiv(4L * 64 * 4096, 256), 256, 0, stream>>>(x, w1, b1, ws + P_HA);

  // ---- stage 2: S2 FFT -> xs (P_XS, planes 65536) ----
  dft_alpha<<<cdiv(4L * 64 * 64 * 31, 256), 256, 0, stream>>>(ws + P_HA, ws + P_PSI,
                                                              ws + P_PSI + 507904);
  s2_spec<<<cdiv(4L * 256 * 64, 256), 256, 0, stream>>>(ws + P_PSI, ws + P_PSI + 507904,
                                                        ws + O_DC32, ws + O_W32,
                                                        ws + P_XS, ws + P_XS + 65536);

  // ---- S2 conv (k2): per-l WMMA GEMM + basis contraction -> zl1 ----
  for (int l = 0; l < 16; ++l) {
    int P = 2 * l + 1;
    float* U  = ws + P_PSI;
    float* Ui = U + (long)4 * P * 384;
    launch_gemm(ws + P_XS + (long)l * l * 64, ws + P_XS + 65536 + (long)l * l * 64,
                ws + O_K2S, nullptr, U, Ui,
                4 * P, 384, 64, P, 256 * 64, P * 384, 384);
    s2_zl<<<cdiv(4L * P * P * 64, 256), 256, 0, stream>>>(
        U, Ui, (const float2*)(ws + O_BS2) + (long)l * l * 6,
        ws + P_ZL, ws + P_ZL + 1396736, P, 5456, 64, off2(l));
  }
  // ifft -> h2 (PSI slot)  [4,64,32,32,32]
  so3_ifft_bias_relu<<<4 * 64 * 32, 256, 0, stream>>>(ws + P_ZL, ws + P_ZL + 1396736,
                                                      ws + O_D16, b2, ws + P_PSI,
                                                      32, 16, 5456, 64);

  // ---- layer 3: so3_conv k3 (64->128, b=16) ----
  so3_spec<<<4 * 64, 256, 0, stream>>>(ws + P_PSI, ws + O_D16, ws + O_W16,
                                       ws + P_XS, ws + P_XS + 1396736, 32, 16, 5456, 64);
  for (int l = 0; l < 16; ++l) {
    int P = 2 * l + 1; int K = P * 64; int N = P * 128;
    float* Br = ws + P_PSI; float* Bi = Br + (long)K * N;
    psi_build<<<cdiv((long)K * N, 256), 256, 0, stream>>>(
        k3, (const float2*)(ws + O_B1) + (long)off2(l) * 36, SC3, Br, Bi, P, 64, 128);
    launch_gemm(ws + P_XS + (long)off2(l) * 64, ws + P_XS + 1396736 + (long)off2(l) * 64,
                Br, Bi,
                ws + P_ZL + (long)off2(l) * 128, ws + P_ZL + 2793472 + (long)off2(l) * 128,
                4 * P, N, K, P, 5456 * 64, 5456 * 128, P * 128);
  }
  so3_ifft_bias_relu<<<4 * 128 * 32, 256, 0, stream>>>(ws + P_ZL, ws + P_ZL + 2793472,
                                                       ws + O_D16, b3, ws + P_HA,
                                                       32, 16, 5456, 128);

  // ---- layer 4: so3_conv k4 (128->128, b_spec=8, grid 32 -> 16) ----
  so3_spec<<<4 * 128, 256, 0, stream>>>(ws + P_HA, ws + O_D16, ws + O_W16,
                                        ws + P_XS, ws + P_XS + 348160, 32, 8, 680, 128);
  for (int l = 0; l < 8; ++l) {
    int P = 2 * l + 1; int K = P * 128; int N = P * 128;
    float* Br = ws + P_PSI; float* Bi = Br + (long)K * N;
    psi_build<<<cdiv((long)K * N, 256), 256, 0, stream>>>(
        k4, (const float2*)(ws + O_B2) + (long)off2(l) * 36, SC4, Br, Bi, P, 128, 128);
    launch_gemm(ws + P_XS + (long)off2(l) * 128, ws + P_XS + 348160 + (long)off2(l) * 128,
                Br, Bi,
                ws + P_ZL + (long)off2(l) * 128, ws + P_ZL + 348160 + (long)off2(l) * 128,
                4 * P, N, K, P, 680 * 128, 680 * 128, P * 128);
  }
  so3_ifft_bias_relu<<<4 * 128 * 16, 256, 0, stream>>>(ws + P_ZL, ws + P_ZL + 348160,
                                                       ws + O_D8, b4, ws + P_PSI,
                                                       16, 8, 680, 128);

  // ---- layer 5: so3_conv k5 (128->256, b=8, grid 16) ----
  so3_spec<<<4 * 128, 256, 0, stream>>>(ws + P_PSI, ws + O_D8, ws + O_W8,
                                        ws + P_XS, ws + P_XS + 348160, 16, 8, 680, 128);
  for (int l = 0; l < 8; ++l) {
    int P = 2 * l + 1; int K = P * 128; int N = P * 256;
    float* Br = ws + P_PSI; float* Bi = Br + (long)K * N;
    psi_build<<<cdiv((long)K * N, 256), 256, 0, stream>>>(
        k5, (const float2*)(ws + O_B3) + (long)off2(l) * 36, SC5, Br, Bi, P, 128, 256);
    launch_gemm(ws + P_XS + (long)off2(l) * 128, ws + P_XS + 348160 + (long)off2(l) * 128,
                Br, Bi,
                ws + P_ZL + (long)off2(l) * 256, ws + P_ZL + 696320 + (long)off2(l) * 256,
                4 * P, N, K, P, 680 * 128, 680 * 256, P * 256);
  }
  so3_ifft_bias_relu<<<4 * 256 * 16, 256, 0, stream>>>(ws + P_ZL, ws + P_ZL + 696320,
                                                       ws + O_D8, b5, ws + P_HA,
                                                       16, 8, 680, 256);

  // ---- integrate + linear -> out[4] ----
  integrate_linear<<<4, 256, 0, stream>>>(ws + P_HA, ws + O_W8, lw, lb, out);
}